// EdgeAggregate_79800492360363
// MI455X (gfx1250) — compile-verified
//
#include <hip/hip_runtime.h>
#include <hip/hip_bf16.h>

// CDNA5 / gfx1250, wave32. f32 WMMA path: V_WMMA_F32_16X16X4_F32.
typedef __attribute__((ext_vector_type(2))) float v2f;
typedef __attribute__((ext_vector_type(8))) float v8f;

#define BATCH 1024
#define NNODE 400
#define NE    128   // edges (BN channels)
#define DJ    128

static __device__ __forceinline__ v8f wmma_f32(v2f a, v2f b, v8f c) {
    // 8 args: (neg_a, A, neg_b, B, c_mod, C, reuse_a, reuse_b)
    return __builtin_amdgcn_wmma_f32_16x16x4_f32(false, a, false, b, (short)0, c, false, false);
}

// ---------------------------------------------------------------------------
// Kernel 1: per (batch, half) compute shift[b] @ gout[b]  ([128,400]@[400,128])
// and write into X[b][e][half*128 + d].  grid = (B, 2), block = 256 (8 waves).
// ---------------------------------------------------------------------------
#define KC1 40
__global__ __launch_bounds__(256) void k_shift_gemm(
    const float* __restrict__ gout, const float* __restrict__ sshift,
    const float* __restrict__ eshift, float* __restrict__ Xbuf)
{
    const int b    = blockIdx.x;
    const int half = blockIdx.y;
    const float* S = (half == 0 ? sshift : eshift) + (size_t)b * NE * NNODE;
    const float* G = gout + (size_t)b * NNODE * DJ;
    float* Xb      = Xbuf + (size_t)b * NE * 256 + half * 128;

    __shared__ float As[128][KC1 + 2];   // shift rows x k-chunk
    __shared__ float Bs[KC1][128 + 4];   // gout  k-chunk x d

    const int tid  = threadIdx.x;
    const int wave = tid >> 5;
    const int lane = tid & 31;
    const int hi   = lane >> 4;          // half-wave select (K pairs / row+8)
    const int l16  = lane & 15;
    const int e0   = wave * 16;          // this wave's 16-row e band

    v8f acc[8];
    #pragma unroll
    for (int t = 0; t < 8; ++t) acc[t] = (v8f)0.f;

    for (int kc = 0; kc < NNODE; kc += KC1) {
        for (int idx = tid; idx < 128 * KC1; idx += 256) {
            int r = idx / KC1, c = idx % KC1;
            As[r][c] = S[r * NNODE + kc + c];
        }
        for (int idx = tid; idx < KC1 * 128; idx += 256) {
            int r = idx >> 7, c = idx & 127;
            Bs[r][c] = G[(size_t)(kc + r) * DJ + c];
        }
        __syncthreads();
        #pragma unroll
        for (int k = 0; k < KC1; k += 4) {
            v2f a;   // A 16x4 f32: lanes0-15 K=0,1 ; lanes16-31 K=2,3
            a.x = As[e0 + l16][k + 2 * hi];
            a.y = As[e0 + l16][k + 2 * hi + 1];
            #pragma unroll
            for (int t = 0; t < 8; ++t) {
                v2f bb;  // B 4x16: lanes0-15 K=0,1 ; lanes16-31 K=2,3 (N striped)
                bb.x = Bs[k + 2 * hi][t * 16 + l16];
                bb.y = Bs[k + 2 * hi + 1][t * 16 + l16];
                acc[t] = wmma_f32(a, bb, acc[t]);
            }
        }
        __syncthreads();
    }
    // C/D layout: VGPR v -> row e0+hi*8+v, col = tile*16 + l16
    #pragma unroll
    for (int t = 0; t < 8; ++t) {
        int col = t * 16 + l16;
        #pragma unroll
        for (int v = 0; v < 8; ++v)
            Xb[(size_t)(e0 + hi * 8 + v) * 256 + col] = acc[t][v];
    }
}

// ---------------------------------------------------------------------------
// Kernel 2: H1 = X @ W1^T + b1  (K=256), write H1 aliased into first half of
// each X block, emit per-(batch,channel e) sum / sumsq partials for BN1.
// ---------------------------------------------------------------------------
#define KC2 32
__global__ __launch_bounds__(256) void k_linear1(
    float* __restrict__ Xbuf, const float* __restrict__ W1,
    const float* __restrict__ b1, float* __restrict__ psum, float* __restrict__ psq)
{
    const int b = blockIdx.x;
    float* Xb = Xbuf + (size_t)b * 32768;     // [128][256]
    float* Hb = Xb;                           // H1 aliases first 16384 floats

    __shared__ float As[128][KC2 + 2];
    __shared__ float Ws[128][KC2 + 2];

    const int tid = threadIdx.x, wave = tid >> 5, lane = tid & 31;
    const int hi = lane >> 4, l16 = lane & 15, e0 = wave * 16;

    v8f acc[8];
    #pragma unroll
    for (int t = 0; t < 8; ++t) acc[t] = (v8f)0.f;

    for (int kc = 0; kc < 256; kc += KC2) {
        for (int idx = tid; idx < 128 * KC2; idx += 256) {
            int r = idx >> 5, c = idx & 31;
            As[r][c] = Xb[r * 256 + kc + c];
            Ws[r][c] = W1[r * 256 + kc + c];   // W1[d][f]; B[k=f][n=d]=Ws[n][k]
        }
        __syncthreads();
        #pragma unroll
        for (int k = 0; k < KC2; k += 4) {
            v2f a;
            a.x = As[e0 + l16][k + 2 * hi];
            a.y = As[e0 + l16][k + 2 * hi + 1];
            #pragma unroll
            for (int t = 0; t < 8; ++t) {
                v2f bb;
                bb.x = Ws[t * 16 + l16][k + 2 * hi];
                bb.y = Ws[t * 16 + l16][k + 2 * hi + 1];
                acc[t] = wmma_f32(a, bb, acc[t]);
            }
        }
        __syncthreads();
    }
    #pragma unroll
    for (int t = 0; t < 8; ++t) {               // bias along d
        float bv = b1[t * 16 + l16];
        #pragma unroll
        for (int v = 0; v < 8; ++v) acc[t][v] += bv;
    }
    #pragma unroll
    for (int t = 0; t < 8; ++t) {               // store H1 [128][128]
        int col = t * 16 + l16;
        #pragma unroll
        for (int v = 0; v < 8; ++v)
            Hb[(e0 + hi * 8 + v) * 128 + col] = acc[t][v];
    }
    // BN1 partials: channel = row e; reduce over d (cols) within the block.
    #pragma unroll
    for (int v = 0; v < 8; ++v) {
        float s = 0.f, q = 0.f;
        #pragma unroll
        for (int t = 0; t < 8; ++t) { float x = acc[t][v]; s += x; q += x * x; }
        #pragma unroll
        for (int m = 8; m >= 1; m >>= 1) { s += __shfl_xor(s, m, 32); q += __shfl_xor(q, m, 32); }
        if (l16 == 0) {
            int r = e0 + hi * 8 + v;
            psum[b * 128 + r] = s;
            psq [b * 128 + r] = q;
        }
    }
}

// ---------------------------------------------------------------------------
// Kernel 3/5: deterministic reduce over batches -> per-channel scale/shift.
// ---------------------------------------------------------------------------
__global__ void k_bnstats(const float* __restrict__ psum, const float* __restrict__ psq,
                          const float* __restrict__ gamma, const float* __restrict__ beta,
                          float* __restrict__ ss)
{
    int e = threadIdx.x;   // 128 threads
    float s = 0.f, q = 0.f;
    for (int b = 0; b < BATCH; ++b) { s += psum[(b << 7) + e]; q += psq[(b << 7) + e]; }
    const float inv = 1.0f / ((float)BATCH * 128.0f);
    float mean = s * inv;
    float var  = q * inv - mean * mean;           // biased variance
    float sc   = gamma[e] * rsqrtf(var + 1e-5f);
    ss[e]       = sc;
    ss[128 + e] = beta[e] - mean * sc;
}

// ---------------------------------------------------------------------------
// Kernel 4: y = relu(BN1(H1)); H2 = y @ W2^T + b2 (K=128), in-place over H1,
// emit BN2 partials.
// ---------------------------------------------------------------------------
__global__ __launch_bounds__(256) void k_linear2(
    float* __restrict__ Xbuf, const float* __restrict__ W2,
    const float* __restrict__ b2, const float* __restrict__ ss1,
    float* __restrict__ psum, float* __restrict__ psq)
{
    const int b = blockIdx.x;
    float* Hb = Xbuf + (size_t)b * 32768;   // H1 in, H2 out (in-place)

    __shared__ float As[128][KC2 + 2];
    __shared__ float Ws[128][KC2 + 2];
    __shared__ float sc[128], sh[128];

    const int tid = threadIdx.x, wave = tid >> 5, lane = tid & 31;
    const int hi = lane >> 4, l16 = lane & 15, e0 = wave * 16;

    if (tid < 128) { sc[tid] = ss1[tid]; sh[tid] = ss1[128 + tid]; }
    __syncthreads();

    v8f acc[8];
    #pragma unroll
    for (int t = 0; t < 8; ++t) acc[t] = (v8f)0.f;

    for (int kc = 0; kc < 128; kc += KC2) {
        for (int idx = tid; idx < 128 * KC2; idx += 256) {
            int r = idx >> 5, c = idx & 31;
            As[r][c] = fmaxf(sc[r] * Hb[r * 128 + kc + c] + sh[r], 0.f);
            Ws[r][c] = W2[r * 128 + kc + c];
        }
        __syncthreads();
        #pragma unroll
        for (int k = 0; k < KC2; k += 4) {
            v2f a;
            a.x = As[e0 + l16][k + 2 * hi];
            a.y = As[e0 + l16][k + 2 * hi + 1];
            #pragma unroll
            for (int t = 0; t < 8; ++t) {
                v2f bb;
                bb.x = Ws[t * 16 + l16][k + 2 * hi];
                bb.y = Ws[t * 16 + l16][k + 2 * hi + 1];
                acc[t] = wmma_f32(a, bb, acc[t]);
            }
        }
        __syncthreads();
    }
    #pragma unroll
    for (int t = 0; t < 8; ++t) {
        float bv = b2[t * 16 + l16];
        #pragma unroll
        for (int v = 0; v < 8; ++v) acc[t][v] += bv;
    }
    #pragma unroll
    for (int t = 0; t < 8; ++t) {
        int col = t * 16 + l16;
        #pragma unroll
        for (int v = 0; v < 8; ++v)
            Hb[(e0 + hi * 8 + v) * 128 + col] = acc[t][v];
    }
    #pragma unroll
    for (int v = 0; v < 8; ++v) {
        float s = 0.f, q = 0.f;
        #pragma unroll
        for (int t = 0; t < 8; ++t) { float x = acc[t][v]; s += x; q += x * x; }
        #pragma unroll
        for (int m = 8; m >= 1; m >>= 1) { s += __shfl_xor(s, m, 32); q += __shfl_xor(q, m, 32); }
        if (l16 == 0) {
            int r = e0 + hi * 8 + v;
            psum[b * 128 + r] = s;
            psq [b * 128 + r] = q;
        }
    }
}

// ---------------------------------------------------------------------------
// Kernel 6: out = relu(eout + scale2[e]*H2 + shift2[e])
// ---------------------------------------------------------------------------
__global__ __launch_bounds__(256) void k_final(
    const float* __restrict__ eout, const float* __restrict__ Xbuf,
    const float* __restrict__ ss2, float* __restrict__ out)
{
    size_t i = (size_t)blockIdx.x * 256 + threadIdx.x;   // 1024*128*128 total
    int e    = (int)((i >> 7) & 127);
    size_t b = i >> 14;
    size_t h = b * 32768 + (i & 16383);                  // H2 aliased layout
    float v  = eout[i] + ss2[e] * Xbuf[h] + ss2[128 + e];
    out[i]   = fmaxf(v, 0.f);
}

extern "C" void kernel_launch(void* const* d_in, const int* in_sizes, int n_in,
                              void* d_out, int out_size, void* d_ws, size_t ws_size,
                              hipStream_t stream) {
    const float* gout   = (const float*)d_in[0];
    const float* eout   = (const float*)d_in[1];
    const float* sshift = (const float*)d_in[2];
    const float* eshift = (const float*)d_in[3];
    const float* W1     = (const float*)d_in[4];
    const float* b1     = (const float*)d_in[5];
    const float* W2     = (const float*)d_in[6];
    const float* b2     = (const float*)d_in[7];
    const float* g1     = (const float*)d_in[8];
    const float* be1    = (const float*)d_in[9];
    const float* g2     = (const float*)d_in[10];
    const float* be2    = (const float*)d_in[11];

    float* ws   = (float*)d_ws;
    float* Xbuf = ws;                                  // B*32768 floats (X; H aliased)
    float* psum = ws + (size_t)BATCH * 32768;          // B*128
    float* psq  = psum + BATCH * 128;                  // B*128
    float* ss1  = psq  + BATCH * 128;                  // 256
    float* ss2  = ss1 + 256;                           // 256
    float* out  = (float*)d_out;

    k_shift_gemm<<<dim3(BATCH, 2), 256, 0, stream>>>(gout, sshift, eshift, Xbuf);
    k_linear1  <<<BATCH, 256, 0, stream>>>(Xbuf, W1, b1, psum, psq);
    k_bnstats  <<<1, 128, 0, stream>>>(psum, psq, g1, be1, ss1);
    k_linear2  <<<BATCH, 256, 0, stream>>>(Xbuf, W2, b2, ss1, psum, psq);
    k_bnstats  <<<1, 128, 0, stream>>>(psum, psq, g2, be2, ss2);
    k_final    <<<(BATCH * NE * 128) / 256, 256, 0, stream>>>(eout, Xbuf, ss2, out);
}